// DetectionLoss_28381143892023
// MI455X (gfx1250) — compile-verified
//
#include <hip/hip_runtime.h>
#include <math.h>

// ---------------------------------------------------------------------------
// YOLO-style detection loss for MI455X (gfx1250, wave32).
// Memory-bound pipeline (~1 GB traffic => ~45us @ 23.3 TB/s). Final scalar
// reductions use V_WMMA_F32_16X16X4_F32 with an all-ones B matrix: exact f32
// wave reduction (rows of D = pair sums of the 32 lane partials), then one
// cross-half shfl_xor(16). EXEC is all-ones at every WMMA call site (grids
// are exact multiples of the block size; no early returns in those kernels).
// ---------------------------------------------------------------------------

#define DEV __device__ __forceinline__

constexpr int   B  = 16;
constexpr int   N  = 33600;   // 160^2 + 80^2 + 40^2
constexpr int   NC = 80;
constexpr int   M  = 32;
constexpr int   RM = 16;      // REG_MAX
constexpr float EPSA = 1e-9f;
constexpr float CLS_GAIN = 0.5f, BOX_GAIN = 7.5f, DFL_GAIN = 1.5f;

// workspace layout (float units)
constexpr size_t OFF_ALIGN = 0;
constexpr size_t OFF_OV    = OFF_ALIGN + (size_t)B * M * N;   // 17.2M floats
constexpr size_t OFF_TH    = OFF_OV    + (size_t)B * M * N;
constexpr size_t OFF_PA    = OFF_TH + (size_t)B * M;
constexpr size_t OFF_PO    = OFF_PA + (size_t)B * M;
constexpr size_t OFF_BITS  = OFF_PO + (size_t)B * M;
constexpr size_t OFF_TLAB  = OFF_BITS + (size_t)B * N;
constexpr size_t OFF_TSC   = OFF_TLAB + (size_t)B * N;
constexpr size_t OFF_ACC   = OFF_TSC  + (size_t)B * N;        // 4 accumulators

typedef __attribute__((ext_vector_type(2))) float v2f;
typedef __attribute__((ext_vector_type(8))) float v8f;

// ------------------------- CIoU (matches reference) ------------------------
DEV float ciou(float x11, float y11, float x12, float y12,
               float x21, float y21, float x22, float y22) {
  const float eps = 1e-7f;
  float w1 = x12 - x11, h1 = y12 - y11;
  float w2 = x22 - x21, h2 = y22 - y21;
  float iw = fmaxf(fminf(x12, x22) - fmaxf(x11, x21), 0.f);
  float ih = fmaxf(fminf(y12, y22) - fmaxf(y11, y21), 0.f);
  float inter = iw * ih;
  float uni = w1 * h1 + w2 * h2 - inter + eps;
  float iou = inter / uni;
  float cw = fmaxf(x12, x22) - fminf(x11, x21);
  float ch = fmaxf(y12, y22) - fminf(y11, y21);
  float c2 = cw * cw + ch * ch + eps;
  float dx = x21 + x22 - x11 - x12;
  float dy = y21 + y22 - y11 - y12;
  float rho2 = (dx * dx + dy * dy) * 0.25f;
  float dv = atanf(w2 / (h2 + eps)) - atanf(w1 / (h1 + eps));
  float v = (4.0f / 9.8696044010893586f) * dv * dv;   // 4/pi^2
  float alpha = v / (v - iou + (1.0f + eps));         // stop-grad value
  return iou - (rho2 / c2 + v * alpha);
}

// -------------- exact f32 wave reduction via V_WMMA_F32_16X16X4_F32 --------
DEV float waveReduceWmma(float s) {
  v2f a; a[0] = s; a[1] = 0.0f;            // A[M=lane%16][K] per A-layout
  v2f bo; bo[0] = 1.0f; bo[1] = 1.0f;      // B = all ones (4x16)
  v8f c = {};
  // D[Mrow][*] = s_row + s_row+16 ; rows 0..7 in c[0..7] (lanes 0-15),
  // rows 8..15 (lanes 16-31).
  c = __builtin_amdgcn_wmma_f32_16x16x4_f32(false, a, false, bo,
                                            (short)0, c, false, false);
  float t = c[0] + c[1] + c[2] + c[3] + c[4] + c[5] + c[6] + c[7];
  t += __shfl_xor(t, 16, 32);              // combine the two half-rows
  return t;                                // full wave sum, all lanes
}

DEV void blockReduceAtomic(float v, float* target, float* smem) {
  float w = waveReduceWmma(v);
  int lane = threadIdx.x & 31;
  int wid  = threadIdx.x >> 5;
  if (lane == 0) smem[wid] = w;
  __syncthreads();
  if (threadIdx.x == 0) {
    float s = 0.f;
    int nw = blockDim.x >> 5;
    for (int i = 0; i < nw; ++i) s += smem[i];
    atomicAdd(target, s);
  }
  __syncthreads();   // allow smem reuse across calls
}

// ------------------------------- kernels -----------------------------------
__global__ void k_zero(float* acc) {
  if (threadIdx.x < 4) acc[threadIdx.x] = 0.0f;
}

// align[b,m,n], overlaps[b,m,n]
__global__ void k_metrics(const float* __restrict__ cls,
                          const float* __restrict__ boxd,
                          const float* __restrict__ tgt,
                          const float* __restrict__ anc,
                          float* __restrict__ alignW,
                          float* __restrict__ ovW) {
  size_t i = (size_t)blockIdx.x * blockDim.x + threadIdx.x;  // exact grid
  int n = (int)(i % N);
  size_t bm = i / N;
  int m = (int)(bm % M);
  int b = (int)(bm / M);
  const float* t = tgt + ((size_t)b * M + m) * 5;
  float lab = t[0], x1 = t[1], y1 = t[2], x2 = t[3], y2 = t[4];
  float area = fmaxf(x2 - x1, 0.f) * fmaxf(y2 - y1, 0.f);
  bool gtm = (lab >= 0.f) && (area > 0.f);
  float ax = anc[2 * n], ay = anc[2 * n + 1];
  float din = fminf(fminf(ax - x1, ay - y1), fminf(x2 - ax, y2 - ay));
  bool inside = din > EPSA;
  const float* pb = boxd + ((size_t)b * N + n) * 4;
  float ov = fmaxf(ciou(x1, y1, x2, y2, pb[0], pb[1], pb[2], pb[3]), 0.f);
  int li = (int)fmaxf(lab, 0.f);
  float x = cls[((size_t)b * N + n) * NC + li];
  float s = 1.0f / (1.0f + expf(-x));
  float o2 = ov * ov;
  alignW[i] = (inside && gtm) ? sqrtf(s) * (o2 * o2 * o2) : 0.0f;
  ovW[i] = ov;
}

// per-(b,m) 10th-largest align value (top-k threshold)
__global__ void k_topk(const float* __restrict__ alignW,
                       float* __restrict__ thresh) {
  int bm = blockIdx.x;
  const float* a = alignW + (size_t)bm * N;
  float loc[10];
#pragma unroll
  for (int k = 0; k < 10; ++k) loc[k] = -1.0f;
  for (int n = threadIdx.x; n < N; n += blockDim.x) {
    float v = a[n];
    if (v > loc[9]) {
      int k = 9;
      while (k > 0 && v > loc[k - 1]) { loc[k] = loc[k - 1]; --k; }
      loc[k] = v;
    }
  }
  __shared__ float sm[256 * 10];
#pragma unroll
  for (int k = 0; k < 10; ++k) sm[threadIdx.x * 10 + k] = loc[k];
  __syncthreads();
  if (threadIdx.x == 0) {
    float best[10];
#pragma unroll
    for (int k = 0; k < 10; ++k) best[k] = -1.0f;
    for (int j = 0; j < 256 * 10; ++j) {
      float v = sm[j];
      if (v > best[9]) {
        int k = 9;
        while (k > 0 && v > best[k - 1]) { best[k] = best[k - 1]; --k; }
        best[k] = v;
      }
    }
    thresh[bm] = best[9];
  }
}

// per-anchor mask bitmask with multi-GT resolution (M=32 -> uint32)
__global__ void k_mask(const float* __restrict__ alignW,
                       const float* __restrict__ ovW,
                       const float* __restrict__ thresh,
                       unsigned* __restrict__ bits) {
  size_t i = (size_t)blockIdx.x * blockDim.x + threadIdx.x;  // exact grid
  int b = (int)(i / N);
  int n = (int)(i % N);
  unsigned bt = 0; int cnt = 0;
  float bo = -1.0f; int bestm = 0;
  for (int m = 0; m < M; ++m) {
    size_t idx = ((size_t)b * M + m) * N + n;
    float al = alignW[idx];
    float ov = ovW[idx];
    if (al >= thresh[b * M + m] && al > EPSA) { bt |= (1u << m); ++cnt; }
    if (ov > bo) { bo = ov; bestm = m; }     // first-max (jnp.argmax)
  }
  if (cnt > 1) bt = (1u << bestm);
  bits[i] = bt;
}

// pos_align[b,m] = max_n align*mp ; pos_ov[b,m] = max_n ov*mp
__global__ void k_posmax(const float* __restrict__ alignW,
                         const float* __restrict__ ovW,
                         const unsigned* __restrict__ bits,
                         float* __restrict__ pa, float* __restrict__ po) {
  int bm = blockIdx.x;
  int b = bm / M, m = bm % M;
  unsigned mbit = 1u << m;
  float ma = 0.f, mo = 0.f;
  for (int n = threadIdx.x; n < N; n += blockDim.x) {
    if (bits[(size_t)b * N + n] & mbit) {
      size_t idx = (size_t)bm * N + n;
      ma = fmaxf(ma, alignW[idx]);
      mo = fmaxf(mo, ovW[idx]);
    }
  }
  __shared__ float s1[256], s2[256];
  s1[threadIdx.x] = ma; s2[threadIdx.x] = mo;
  __syncthreads();
  for (int st = 128; st > 0; st >>= 1) {
    if (threadIdx.x < (unsigned)st) {
      s1[threadIdx.x] = fmaxf(s1[threadIdx.x], s1[threadIdx.x + st]);
      s2[threadIdx.x] = fmaxf(s2[threadIdx.x], s2[threadIdx.x + st]);
    }
    __syncthreads();
  }
  if (threadIdx.x == 0) { pa[bm] = s1[0]; po[bm] = s2[0]; }
}

// per-anchor: norm, target gather, CIoU + DFL contributions; tss
__global__ void k_peranchor(const float* __restrict__ boxd,
                            const float* __restrict__ dist,
                            const float* __restrict__ tgt,
                            const float* __restrict__ anc,
                            const float* __restrict__ strd,
                            const float* __restrict__ alignW,
                            const unsigned* __restrict__ bits,
                            const float* __restrict__ pa,
                            const float* __restrict__ po,
                            int* __restrict__ tlab,
                            float* __restrict__ tsc,
                            float* __restrict__ acc) {
  size_t i = (size_t)blockIdx.x * blockDim.x + threadIdx.x;  // exact grid
  int b = (int)(i / N);
  int n = (int)(i % N);
  unsigned bt = bits[i];
  bool fg = bt != 0;
  int gidx = fg ? (__ffs(bt) - 1) : 0;
  float norm = 0.0f;
  unsigned tb = bt;
  while (tb) {
    int m = __ffs(tb) - 1; tb &= tb - 1;
    float am = alignW[((size_t)b * M + m) * N + n];
    norm = fmaxf(norm, am * po[b * M + m] / (pa[b * M + m] + EPSA));
  }
  const float* t = tgt + ((size_t)b * M + gidx) * 5;
  int lab = (int)fmaxf(t[0], 0.f);
  float tx1 = t[1], ty1 = t[2], tx2 = t[3], ty2 = t[4];
  float tscale = fg ? norm : 0.0f;
  tlab[i] = lab;
  tsc[i]  = tscale;
  float weight = tscale;
  float c_iou = 0.0f, c_dfl = 0.0f;
  if (weight > 0.0f) {
    const float* pb = boxd + i * 4;
    float iou = ciou(pb[0], pb[1], pb[2], pb[3], tx1, ty1, tx2, ty2);
    c_iou = (1.0f - iou) * weight;
    float ax = anc[2 * n], ay = anc[2 * n + 1], st = strd[n];
    float ltrb[4] = { (ax - tx1) / st, (ay - ty1) / st,
                      (tx2 - ax) / st, (ty2 - ay) / st };
    float dfl = 0.0f;
#pragma unroll
    for (int sd = 0; sd < 4; ++sd) {
      float tv = fminf(fmaxf(ltrb[sd], 0.0f), (float)RM - 1.0f - 0.01f);
      int   tl = (int)tv;
      float wl = (float)(tl + 1) - tv;
      const float* x = dist + (i * 4 + sd) * RM;
      float mx = x[0];
      for (int j = 1; j < RM; ++j) mx = fmaxf(mx, x[j]);
      float se = 0.f;
      for (int j = 0; j < RM; ++j) se += expf(x[j] - mx);
      float lse = mx + logf(se);
      float ll = x[tl] - lse, lr = x[tl + 1] - lse;
      dfl += -(ll * wl + lr * (1.0f - wl));
    }
    c_dfl = dfl * 0.25f * weight;
  }
  __shared__ float smem[8];
  blockReduceAtomic(weight, acc + 0, smem);   // tss raw
  blockReduceAtomic(c_iou,  acc + 2, smem);
  blockReduceAtomic(c_dfl,  acc + 3, smem);
}

// streaming BCE over B*N*NC logits, float4 per thread (80 % 4 == 0)
__global__ void k_cls(const float* __restrict__ cls,
                      const int* __restrict__ tlab,
                      const float* __restrict__ tsc,
                      float* __restrict__ acc) {
  size_t i = (size_t)blockIdx.x * blockDim.x + threadIdx.x;  // exact grid
  const float4* c4 = (const float4*)cls;
  float4 x4 = c4[i];
  size_t e  = i * 4;
  size_t bn = e / NC;
  int    c0 = (int)(e % NC);
  int    lab = tlab[bn];
  float  ts  = tsc[bn];
  float xs[4] = { x4.x, x4.y, x4.z, x4.w };
  float sum = 0.0f;
#pragma unroll
  for (int j = 0; j < 4; ++j) {
    float x  = xs[j];
    float te = (c0 + j == lab) ? ts : 0.0f;
    bool  pos = te > 0.0f;
    float p  = 1.0f / (1.0f + expf(-x));
    float w  = pos ? te : 0.75f * p * p;
    float bce = fmaxf(x, 0.0f) - x * te + log1pf(expf(-fabsf(x)));
    sum += bce * w;
  }
  __shared__ float smem[8];
  blockReduceAtomic(sum, acc + 1, smem);
}

__global__ void k_final(const float* __restrict__ acc, float* __restrict__ out) {
  if (threadIdx.x == 0) {
    float tss = fmaxf(acc[0], 1.0f);
    float lc = CLS_GAIN * acc[1] / tss;
    float li = BOX_GAIN * acc[2] / tss;
    float ld = DFL_GAIN * acc[3] / tss;
    out[0] = (lc + li + ld) * (float)B;
    out[1] = lc; out[2] = li; out[3] = ld;
  }
}

// ------------------------------- launcher ----------------------------------
extern "C" void kernel_launch(void* const* d_in, const int* in_sizes, int n_in,
                              void* d_out, int out_size, void* d_ws, size_t ws_size,
                              hipStream_t stream) {
  (void)in_sizes; (void)n_in; (void)out_size; (void)ws_size;
  const float* cls  = (const float*)d_in[0];   // (B,N,NC)
  const float* dist = (const float*)d_in[1];   // (B,N,4,RM)
  const float* boxd = (const float*)d_in[2];   // (B,N,4)
  const float* tgt  = (const float*)d_in[3];   // (B,M,5)
  const float* anc  = (const float*)d_in[4];   // (N,2)
  const float* strd = (const float*)d_in[5];   // (N,1)

  float*    ws     = (float*)d_ws;
  float*    alignW = ws + OFF_ALIGN;
  float*    ovW    = ws + OFF_OV;
  float*    thresh = ws + OFF_TH;
  float*    pa     = ws + OFF_PA;
  float*    po     = ws + OFF_PO;
  unsigned* bits   = (unsigned*)(ws + OFF_BITS);
  int*      tlab   = (int*)(ws + OFF_TLAB);
  float*    tsc    = ws + OFF_TSC;
  float*    acc    = ws + OFF_ACC;
  float*    out    = (float*)d_out;

  k_zero<<<1, 64, 0, stream>>>(acc);

  size_t tot_bmn = (size_t)B * M * N;                 // 17,203,200 = 256*67200
  k_metrics<<<(unsigned)(tot_bmn / 256), 256, 0, stream>>>(cls, boxd, tgt, anc,
                                                           alignW, ovW);
  k_topk<<<B * M, 256, 0, stream>>>(alignW, thresh);

  size_t tot_bn = (size_t)B * N;                      // 537,600 = 256*2100
  k_mask<<<(unsigned)(tot_bn / 256), 256, 0, stream>>>(alignW, ovW, thresh, bits);
  k_posmax<<<B * M, 256, 0, stream>>>(alignW, ovW, bits, pa, po);
  k_peranchor<<<(unsigned)(tot_bn / 256), 256, 0, stream>>>(
      boxd, dist, tgt, anc, strd, alignW, bits, pa, po, tlab, tsc, acc);

  size_t tot_c4 = (size_t)B * N * NC / 4;             // 10,752,000 = 256*42000
  k_cls<<<(unsigned)(tot_c4 / 256), 256, 0, stream>>>(cls, tlab, tsc, acc);

  k_final<<<1, 64, 0, stream>>>(acc, out);
}